// GATRefiner_18786186952802
// MI455X (gfx1250) — compile-verified
//
#include <hip/hip_runtime.h>

typedef float v2f __attribute__((ext_vector_type(2)));
typedef float v8f __attribute__((ext_vector_type(8)));

#define NEG_SLOPE 0.2f

// ---------------------------------------------------------------------------
// FP32 WMMA GEMM: C[M,N] = A[M,K] x B[K,N]   (row-major, M%16==0, N%64==0,
// K%4==0).  One wave32 per 16x64 output strip: 4 accumulators share one A
// fragment per K-step -> 4x V_WMMA_F32_16X16X4_F32 per iteration.
// A frag (16x4, ISA 7.12.2): lane L holds row M=L%16; lanes<16 carry K={0,1}
// in v[0],v[1]; lanes>=16 carry K={2,3}.  B frag mirrors (row per VGPR,
// striped across lanes).  C/D: VGPR r -> row r + 8*(lane>=16), col = lane%16.
// ---------------------------------------------------------------------------
__global__ __launch_bounds__(32)
void gat_gemm_wmma(const float* __restrict__ A, const float* __restrict__ B,
                   float* __restrict__ C, int M, int K, int N) {
  const int n0   = blockIdx.x << 6;      // 64-wide N strip
  const int m0   = blockIdx.y << 4;
  const int lane = threadIdx.x;          // 0..31
  const int half = lane >> 4;            // 0 or 1
  const int l16  = lane & 15;
  const int kk   = half << 1;            // K sub-offset {0,2}
  const float* arow = A + (size_t)(m0 + l16) * K + kk;
  const float* brow = B + (size_t)kk * N + n0 + l16;
  v8f acc0 = {}, acc1 = {}, acc2 = {}, acc3 = {};
  for (int k0 = 0; k0 < K; k0 += 4) {
    if ((k0 & 28) == 0)  // prefetch next A cacheline chunk (global_prefetch_b8)
      __builtin_prefetch(arow + k0 + 32, 0, 0);
    v2f a;
    a.x = arow[k0];
    a.y = arow[k0 + 1];
    const float* b0 = brow + (size_t)k0 * N;   // row  k0+kk
    const float* b1 = b0 + N;                  // row  k0+kk+1
    v2f bb0, bb1, bb2, bb3;
    bb0.x = b0[0];  bb0.y = b1[0];
    bb1.x = b0[16]; bb1.y = b1[16];
    bb2.x = b0[32]; bb2.y = b1[32];
    bb3.x = b0[48]; bb3.y = b1[48];
    acc0 = __builtin_amdgcn_wmma_f32_16x16x4_f32(false, a, false, bb0, (short)0,
                                                 acc0, false, false);
    acc1 = __builtin_amdgcn_wmma_f32_16x16x4_f32(false, a, false, bb1, (short)0,
                                                 acc1, false, false);
    acc2 = __builtin_amdgcn_wmma_f32_16x16x4_f32(false, a, false, bb2, (short)0,
                                                 acc2, false, false);
    acc3 = __builtin_amdgcn_wmma_f32_16x16x4_f32(false, a, false, bb3, (short)0,
                                                 acc3, false, false);
  }
  float* crow = C + (size_t)(m0 + (half << 3)) * N + n0 + l16;
#pragma unroll
  for (int r = 0; r < 8; ++r) {
    crow[(size_t)r * N + 0]  = acc0[r];
    crow[(size_t)r * N + 16] = acc1[r];
    crow[(size_t)r * N + 32] = acc2[r];
    crow[(size_t)r * N + 48] = acc3[r];
  }
}

// ---------------------------------------------------------------------------
__global__ void gat_fill_f32(float* __restrict__ p, float v, int n) {
  int t = blockIdx.x * blockDim.x + threadIdx.x;
  if (t < n) p[t] = v;
}

__global__ void gat_copy_f32(const float* __restrict__ in, float* __restrict__ out,
                             int n) {
  int t = blockIdx.x * blockDim.x + threadIdx.x;
  if (t < n) out[t] = in[t];
}

// e_s[n,h] = sum_c h[n,h,c]*a_src[h,c];  e_d likewise.  One thread per (n,h).
__global__ void gat_attn_logits(const float* __restrict__ h,
                                const float* __restrict__ a_src,
                                const float* __restrict__ a_dst,
                                float* __restrict__ e_s, float* __restrict__ e_d,
                                int Nn, int H, int C) {
  int t = blockIdx.x * blockDim.x + threadIdx.x;
  if (t >= Nn * H) return;
  int n = t / H, hh = t % H;
  const float* hp = h + (size_t)n * H * C + (size_t)hh * C;
  const float* as = a_src + (size_t)hh * C;
  const float* ad = a_dst + (size_t)hh * C;
  float ss = 0.f, sd = 0.f;
  for (int c = 0; c < C; ++c) {
    float v = hp[c];
    ss += v * as[c];
    sd += v * ad[c];
  }
  e_s[t] = ss;
  e_d[t] = sd;
}

// float atomic max via signed/unsigned int monotonicity trick
__device__ __forceinline__ void gat_atomic_max_f32(float* addr, float val) {
  if (val >= 0.0f)
    atomicMax((int*)addr, __float_as_int(val));
  else
    atomicMin((unsigned int*)addr, __float_as_uint(val));
}

__device__ __forceinline__ void gat_edge_endpoints(const int* __restrict__ ei,
                                                   int E, int i, int& src,
                                                   int& dst) {
  if (i < E) {
    src = ei[i];
    dst = ei[E + i];
  } else {             // implicit self-loop appended after the real edges
    src = i - E;
    dst = i - E;
  }
}

// Pass 1: segment max of leaky_relu(e_s[src]+e_d[dst]) over dst.
__global__ void gat_edge_max(const int* __restrict__ ei, int E, int Nn,
                             const float* __restrict__ e_s,
                             const float* __restrict__ e_d,
                             float* __restrict__ m, int H) {
  int i = blockIdx.x * blockDim.x + threadIdx.x;
  if (i >= E + Nn) return;
  int src, dst;
  gat_edge_endpoints(ei, E, i, src, dst);
  for (int h = 0; h < H; ++h) {
    float e = e_s[src * H + h] + e_d[dst * H + h];
    e = (e > 0.f) ? e : NEG_SLOPE * e;
    gat_atomic_max_f32(&m[dst * H + h], e);
  }
}

// Pass 2: p = exp(e - m[dst]); segment-sum p over dst; stash p for pass 3.
__global__ void gat_edge_expsum(const int* __restrict__ ei, int E, int Nn,
                                const float* __restrict__ e_s,
                                const float* __restrict__ e_d,
                                const float* __restrict__ m,
                                float* __restrict__ s, float* __restrict__ p,
                                int H) {
  int i = blockIdx.x * blockDim.x + threadIdx.x;
  if (i >= E + Nn) return;
  int src, dst;
  gat_edge_endpoints(ei, E, i, src, dst);
  for (int h = 0; h < H; ++h) {
    float e = e_s[src * H + h] + e_d[dst * H + h];
    e = (e > 0.f) ? e : NEG_SLOPE * e;
    float pv = expf(e - m[dst * H + h]);
    p[(size_t)i * H + h] = pv;
    atomicAdd(&s[dst * H + h], pv);
  }
}

// Pass 2.5: normalize p -> alpha = p / (s[dst] + eps)  (in place on p).
__global__ void gat_alpha(const int* __restrict__ ei, int E, int Nn,
                          const float* __restrict__ s, float* __restrict__ p,
                          int H) {
  int t = blockIdx.x * blockDim.x + threadIdx.x;
  if (t >= (E + Nn) * H) return;
  int i = t / H, hh = t % H;
  int src, dst;
  gat_edge_endpoints(ei, E, i, src, dst);
  p[t] = p[t] / (s[dst * H + hh] + 1e-16f);
}

// Pass 3: out[dst,h,c] += alpha[i,h] * h[src,h,c].  One thread per
// (edge, 4-channel group): float4 gathers of h[src] (global_load_b128),
// src/dst loads broadcast within the 64 threads sharing an edge.
__global__ void gat_edge_scatter(const int* __restrict__ ei, int E, int Nn,
                                 const float* __restrict__ h,
                                 const float* __restrict__ alpha,
                                 float* __restrict__ agg, int H, int C) {
  const int HC = H * C;
  const int G = HC >> 2;  // float4 groups per edge
  long long t = (long long)blockIdx.x * blockDim.x + threadIdx.x;
  long long total = (long long)(E + Nn) * G;
  if (t >= total) return;
  int i = (int)(t / G);
  int ch = ((int)(t % G)) << 2;
  int hh = ch / C;  // C%4==0 -> group stays inside one head
  int src, dst;
  gat_edge_endpoints(ei, E, i, src, dst);
  float al = alpha[(size_t)i * H + hh];
  const float4 hv = *(const float4*)(h + (size_t)src * HC + ch);
  float* ap = agg + (size_t)dst * HC + ch;
  atomicAdd(ap + 0, al * hv.x);
  atomicAdd(ap + 1, al * hv.y);
  atomicAdd(ap + 2, al * hv.z);
  atomicAdd(ap + 3, al * hv.w);
}

// out = (elu?) (agg + bias[c])
__global__ void gat_bias_act(float* __restrict__ agg, const float* __restrict__ b,
                             int Nn, int F, int do_elu) {
  int t = blockIdx.x * blockDim.x + threadIdx.x;
  if (t >= Nn * F) return;
  float v = agg[t] + b[t % F];
  if (do_elu) v = (v > 0.f) ? v : (expf(v) - 1.0f);
  agg[t] = v;
}

// node_output[n,0:2] = h3[n,:] @ cls_W + cls_b
__global__ void gat_cls_head(const float* __restrict__ h3,
                             const float* __restrict__ W,
                             const float* __restrict__ b,
                             float* __restrict__ out, int Nn, int C) {
  int n = blockIdx.x * blockDim.x + threadIdx.x;
  if (n >= Nn) return;
  const float* hp = h3 + (size_t)n * C;
  float o0 = b[0], o1 = b[1];
  for (int c = 0; c < C; ++c) {
    float v = hp[c];
    o0 += v * W[c * 2 + 0];
    o1 += v * W[c * 2 + 1];
  }
  out[n * 2 + 0] = o0;
  out[n * 2 + 1] = o1;
}

// ---------------------------------------------------------------------------
extern "C" void kernel_launch(void* const* d_in, const int* in_sizes, int n_in,
                              void* d_out, int out_size, void* d_ws,
                              size_t ws_size, hipStream_t stream) {
  const float* x   = (const float*)d_in[0];
  const int*   ei  = (const int*)d_in[1];
  const float* W1  = (const float*)d_in[2];
  const float* a1s = (const float*)d_in[3];
  const float* a1d = (const float*)d_in[4];
  const float* b1  = (const float*)d_in[5];
  const float* W2  = (const float*)d_in[6];
  const float* a2s = (const float*)d_in[7];
  const float* a2d = (const float*)d_in[8];
  const float* b2  = (const float*)d_in[9];
  const float* W3  = (const float*)d_in[10];
  const float* a3s = (const float*)d_in[11];
  const float* a3d = (const float*)d_in[12];
  const float* b3  = (const float*)d_in[13];
  const float* cW  = (const float*)d_in[14];
  const float* cb  = (const float*)d_in[15];

  const int Fin = 128, H = 4, Cc = 64, HC = H * Cc;  // 256
  const int N  = in_sizes[0] / Fin;                  // 50000
  const int E  = in_sizes[1] / 2;                    // 800000
  const int Et = E + N;                              // edges + self loops

  // workspace carve-up (floats)
  float* ws   = (float*)d_ws;
  float* hbuf = ws;                         // N*HC : transformed features h
  float* agg  = hbuf + (size_t)N * HC;      // N*HC : aggregated output
  float* es   = agg + (size_t)N * HC;       // N*H
  float* ed   = es + (size_t)N * H;         // N*H
  float* mbuf = ed + (size_t)N * H;         // N*H
  float* sbuf = mbuf + (size_t)N * H;       // N*H
  float* pbuf = sbuf + (size_t)N * H;       // Et*H

  const int TPB = 256;
  const float NEG_INF = -__builtin_inff();

  auto gblk = [&](long long n) { return (unsigned int)((n + TPB - 1) / TPB); };

  // -------- run one GAT conv's softmax-aggregate given hbuf (h = in @ W) ----
  auto run_edges = [&](const float* aS, const float* aD, int Hh) {
    const int HCc = Hh * Cc;
    gat_attn_logits<<<gblk((long long)N * Hh), TPB, 0, stream>>>(hbuf, aS, aD,
                                                                 es, ed, N, Hh, Cc);
    gat_fill_f32<<<gblk((long long)N * Hh), TPB, 0, stream>>>(mbuf, NEG_INF, N * Hh);
    gat_fill_f32<<<gblk((long long)N * Hh), TPB, 0, stream>>>(sbuf, 0.f, N * Hh);
    gat_fill_f32<<<gblk((long long)N * HCc), TPB, 0, stream>>>(agg, 0.f, N * HCc);
    gat_edge_max<<<gblk(Et), TPB, 0, stream>>>(ei, E, N, es, ed, mbuf, Hh);
    gat_edge_expsum<<<gblk(Et), TPB, 0, stream>>>(ei, E, N, es, ed, mbuf, sbuf,
                                                  pbuf, Hh);
    gat_alpha<<<gblk((long long)Et * Hh), TPB, 0, stream>>>(ei, E, N, sbuf, pbuf, Hh);
    gat_edge_scatter<<<gblk((long long)Et * (HCc >> 2)), TPB, 0, stream>>>(
        ei, E, N, hbuf, pbuf, agg, Hh, Cc);
  };

  // ---------------- layer 1: x[N,128] -> h[N,256], elu ----------------------
  gat_gemm_wmma<<<dim3(HC / 64, N / 16), 32, 0, stream>>>(x, W1, hbuf, N, Fin, HC);
  run_edges(a1s, a1d, H);
  gat_bias_act<<<gblk((long long)N * HC), TPB, 0, stream>>>(agg, b1, N, HC, 1);

  // ---------------- layer 2: h[N,256] -> h[N,256], elu ----------------------
  gat_gemm_wmma<<<dim3(HC / 64, N / 16), 32, 0, stream>>>(agg, W2, hbuf, N, HC, HC);
  run_edges(a2s, a2d, H);
  gat_bias_act<<<gblk((long long)N * HC), TPB, 0, stream>>>(agg, b2, N, HC, 1);

  // ---------------- layer 3: h[N,256] -> h[N,64], H=1, no elu ---------------
  gat_gemm_wmma<<<dim3(Cc / 64, N / 16), 32, 0, stream>>>(agg, W3, hbuf, N, HC, Cc);
  run_edges(a3s, a3d, 1);
  gat_bias_act<<<gblk((long long)N * Cc), TPB, 0, stream>>>(agg, b3, N, Cc, 0);

  // ---------------- heads: node_output [N,2] then link_embedding [N,64] ----
  float* out = (float*)d_out;
  gat_cls_head<<<gblk(N), TPB, 0, stream>>>(agg, cW, cb, out, N, Cc);
  gat_copy_f32<<<gblk((long long)N * Cc), TPB, 0, stream>>>(agg, out + (size_t)2 * N,
                                                            N * Cc);
}